// GraphAttentionLayer_40578851012715
// MI455X (gfx1250) — compile-verified
//
#include <hip/hip_runtime.h>
#include <hip/hip_bf16.h>
#include <math.h>

// Problem constants (match reference)
#define BATCH 2
#define NN    2048
#define DD    128
#define HH    128
#define HQ    32
#define EPS_DIST 1e-8f
#define EPS_LN   1e-5f
#define SCALE    0.08838834764831845f   // 1/sqrt(128)

typedef __attribute__((ext_vector_type(2))) float v2f;
typedef __attribute__((ext_vector_type(8))) float v8f;

// D = A(16x4,f32) * B(4x16,f32) + C(16x16,f32)   -> v_wmma_f32_16x16x4_f32
__device__ __forceinline__ v8f wmma_f32(v2f a, v2f b, v8f c) {
    return __builtin_amdgcn_wmma_f32_16x16x4_f32(
        /*neg_a=*/false, a, /*neg_b=*/false, b,
        /*c_mod=*/(short)0, c, /*reuse_a=*/false, /*reuse_b=*/false);
}

__device__ __forceinline__ float silu(float x) {
    return x / (1.0f + __expf(-x));
}

// ---------------------------------------------------------------------------
// Kernel 1: fused QKV projection.  q/k/v = h @ W + b
// h: (4096,128) row-major, W: (128,128) row-major.  One wave per 16x16 tile,
// A tile (h) loaded once and reused for all three weight matrices.
// ---------------------------------------------------------------------------
__global__ void qkv_gemm_kernel(const float* __restrict__ h,
                                const float* __restrict__ Wq, const float* __restrict__ bq,
                                const float* __restrict__ Wk, const float* __restrict__ bk,
                                const float* __restrict__ Wv, const float* __restrict__ bv,
                                float* __restrict__ q, float* __restrict__ k,
                                float* __restrict__ v) {
    const int lane = threadIdx.x & 31;
    const int wave = threadIdx.x >> 5;
    const int tile = blockIdx.x * 8 + wave;           // 2048 tiles total
    const int mt = tile >> 3;                         // 256 M tiles (rows of 16)
    const int nt = tile & 7;                          // 8 N tiles (cols of 16)
    const int m15 = lane & 15;
    const int half = lane >> 4;                       // 0: K pair {0,1}, 1: {2,3}
    const int arow = mt * 16 + m15;
    const int col  = nt * 16 + m15;

    v8f cq = {}; v8f ck = {}; v8f cv = {};
#pragma unroll
    for (int kk = 0; kk < 128; kk += 4) {
        const int ka = kk + half * 2;
        v2f a = *(const v2f*)(h + arow * DD + ka);    // A: lane=M, 2 consecutive K
        v2f bwq, bwk, bwv;                            // B[k][n] = W[ka][col]
        bwq.x = Wq[ka * HH + col]; bwq.y = Wq[(ka + 1) * HH + col];
        bwk.x = Wk[ka * HH + col]; bwk.y = Wk[(ka + 1) * HH + col];
        bwv.x = Wv[ka * HH + col]; bwv.y = Wv[(ka + 1) * HH + col];
        cq = wmma_f32(a, bwq, cq);
        ck = wmma_f32(a, bwk, ck);
        cv = wmma_f32(a, bwv, cv);
    }
    const float biq = bq[col], bik = bk[col], biv = bv[col];
#pragma unroll
    for (int r = 0; r < 8; ++r) {
        const int row = mt * 16 + r + half * 8;       // C layout: VGPR r -> M=r(+8)
        q[row * HH + col] = cq[r] + biq;
        k[row * HH + col] = ck[r] + bik;
        v[row * HH + col] = cv[r] + biv;
    }
}

// ---------------------------------------------------------------------------
// Kernel 2: attention.  One workgroup (8 waves) owns 16 query rows and ALL
// 2048 keys.  Full 16x2048 score strip lives in LDS (128 KB of the 320 KB WGP
// LDS).  Phase1: WMMA QK^T + distance-MLP bias -> LDS.  Phase2: softmax in
// LDS (wave32 shuffle reductions).  Phase3: WMMA PV reading P from LDS.
// ---------------------------------------------------------------------------
__global__ void attn_kernel(const float* __restrict__ q,
                            const float* __restrict__ kmat,
                            const float* __restrict__ vmat,
                            const float* __restrict__ coords,
                            const float* __restrict__ Wd1, const float* __restrict__ bd1,
                            const float* __restrict__ Wd2, const float* __restrict__ bd2,
                            float* __restrict__ attn_out) {
    __shared__ float Slds[16 * NN];     // 131072 B
    __shared__ float Qlds[16 * HH];     //   8192 B
    __shared__ float qcx[16], qcy[16];
    __shared__ float w1[HQ], b1s[HQ], w2[HQ];
    __shared__ float b2s;
    __shared__ float rowinv[16];

    const int lane = threadIdx.x & 31;
    const int wave = threadIdx.x >> 5;
    const int tid  = threadIdx.x;
    const int b    = blockIdx.x >> 7;          // batch
    const int qt   = blockIdx.x & 127;         // query tile
    const int q0   = qt * 16;
    const int m15  = lane & 15;
    const int half = lane >> 4;
    const int qbase = b * NN + q0;

    // --- stage Q tile, query coords, MLP weights into LDS ---
#pragma unroll
    for (int i = 0; i < 8; ++i)
        Qlds[tid * 8 + i] = q[qbase * HH + tid * 8 + i];
    if (tid < 16) {
        qcx[tid] = coords[(qbase + tid) * 2 + 0];
        qcy[tid] = coords[(qbase + tid) * 2 + 1];
    }
    if (tid < HQ) { w1[tid] = Wd1[tid]; b1s[tid] = bd1[tid]; w2[tid] = Wd2[tid]; }
    if (tid == 0) b2s = bd2[0];
    __syncthreads();

    // --- Phase 1: scores = QK^T * scale + dist_bias, into LDS ---
    for (int kt = wave; kt < 128; kt += 8) {
        const int j0  = kt * 16;
        const int key = b * NN + j0 + m15;
        v8f c = {};
#pragma unroll
        for (int kk = 0; kk < 128; kk += 4) {
            const int ka = kk + half * 2;
            v2f a  = *(const v2f*)(&Qlds[m15 * HH + ka]);     // A: lane=query
            v2f bb = *(const v2f*)(&kmat[key * HH + ka]);     // B[k][n]=K[key_n][k]
            c = wmma_f32(a, bb, c);
        }
        const float kx = coords[key * 2 + 0];
        const float ky = coords[key * 2 + 1];
#pragma unroll
        for (int r = 0; r < 8; ++r) {
            const int m = r + half * 8;
            const float dx = qcx[m] - kx;
            const float dy = qcy[m] - ky;
            const float d  = sqrtf(dx * dx + dy * dy + EPS_DIST);
            float bias = b2s;
#pragma unroll 8
            for (int t = 0; t < HQ; ++t) {
                const float z = fmaf(d, w1[t], b1s[t]);
                bias = fmaf(silu(z), w2[t], bias);
            }
            Slds[m * NN + j0 + m15] = fmaf(c[r], SCALE, bias);
        }
    }
    __syncthreads();

    // --- Phase 2: row softmax in LDS; keep 1/sum for the PV epilogue ---
#pragma unroll
    for (int rr = 0; rr < 2; ++rr) {
        const int m = wave * 2 + rr;
        float mx = -3.402823466e38f;
        for (int j = lane; j < NN; j += 32) mx = fmaxf(mx, Slds[m * NN + j]);
#pragma unroll
        for (int o = 16; o > 0; o >>= 1) mx = fmaxf(mx, __shfl_xor(mx, o, 32));
        float sum = 0.0f;
        for (int j = lane; j < NN; j += 32) {
            const float e = __expf(Slds[m * NN + j] - mx);
            Slds[m * NN + j] = e;
            sum += e;
        }
#pragma unroll
        for (int o = 16; o > 0; o >>= 1) sum += __shfl_xor(sum, o, 32);
        if (lane == 0) rowinv[m] = 1.0f / sum;
    }
    __syncthreads();

    // --- Phase 3: out = P @ V, P streamed from LDS in A layout ---
    const int c0 = wave * 16;                  // each wave owns 16 of H=128 cols
    v8f acc = {};
    for (int kk = 0; kk < NN; kk += 4) {
        const int ka = kk + half * 2;
        v2f a = *(const v2f*)(&Slds[m15 * NN + ka]);          // A: lane=query, K=keys
        const int vrow = b * NN + ka;
        v2f bb;
        bb.x = vmat[vrow * HH + c0 + m15];                    // B[k][n]=V[key][col]
        bb.y = vmat[(vrow + 1) * HH + c0 + m15];
        acc = wmma_f32(a, bb, acc);
    }
#pragma unroll
    for (int r = 0; r < 8; ++r) {
        const int m = r + half * 8;
        attn_out[(qbase + m) * HH + c0 + m15] = acc[r] * rowinv[m];
    }
}

// ---------------------------------------------------------------------------
// Kernel 3: output projection  proj = attn_out @ Wo + bo   (WMMA f32 GEMM)
// ---------------------------------------------------------------------------
__global__ void out_gemm_kernel(const float* __restrict__ a_in,
                                const float* __restrict__ Wo,
                                const float* __restrict__ bo,
                                float* __restrict__ proj) {
    const int lane = threadIdx.x & 31;
    const int wave = threadIdx.x >> 5;
    const int tile = blockIdx.x * 8 + wave;
    const int mt = tile >> 3;
    const int nt = tile & 7;
    const int m15 = lane & 15;
    const int half = lane >> 4;
    const int arow = mt * 16 + m15;
    const int col  = nt * 16 + m15;

    v8f c = {};
#pragma unroll
    for (int kk = 0; kk < 128; kk += 4) {
        const int ka = kk + half * 2;
        v2f a = *(const v2f*)(a_in + arow * HH + ka);
        v2f bb;
        bb.x = Wo[ka * DD + col]; bb.y = Wo[(ka + 1) * DD + col];
        c = wmma_f32(a, bb, c);
    }
    const float bi = bo[col];
#pragma unroll
    for (int r = 0; r < 8; ++r) {
        const int row = mt * 16 + r + half * 8;
        proj[row * DD + col] = c[r] + bi;
    }
}

// ---------------------------------------------------------------------------
// Kernel 4: res = h + silu(proj); LayerNorm(res)*gamma + beta
// One wave32 per row of 128, 4 elements per lane, shuffle reductions.
// ---------------------------------------------------------------------------
__global__ void epilogue_kernel(const float* __restrict__ proj,
                                const float* __restrict__ h,
                                const float* __restrict__ gamma,
                                const float* __restrict__ beta,
                                float* __restrict__ out) {
    const int lane = threadIdx.x & 31;
    const int wave = threadIdx.x >> 5;
    const int row  = blockIdx.x * 8 + wave;        // 4096 rows
    const int base = row * DD + lane * 4;

    float rv[4];
    float s = 0.0f;
#pragma unroll
    for (int i = 0; i < 4; ++i) {
        const float p = proj[base + i];
        rv[i] = h[base + i] + silu(p);
        s += rv[i];
    }
#pragma unroll
    for (int o = 16; o > 0; o >>= 1) s += __shfl_xor(s, o, 32);
    const float mu = s * (1.0f / DD);
    float var = 0.0f;
#pragma unroll
    for (int i = 0; i < 4; ++i) { const float d = rv[i] - mu; var += d * d; }
#pragma unroll
    for (int o = 16; o > 0; o >>= 1) var += __shfl_xor(var, o, 32);
    const float rstd = rsqrtf(var * (1.0f / DD) + EPS_LN);
#pragma unroll
    for (int i = 0; i < 4; ++i) {
        const int d = lane * 4 + i;
        out[base + i] = (rv[i] - mu) * rstd * gamma[d] + beta[d];
    }
}

// ---------------------------------------------------------------------------
extern "C" void kernel_launch(void* const* d_in, const int* in_sizes, int n_in,
                              void* d_out, int out_size, void* d_ws, size_t ws_size,
                              hipStream_t stream) {
    const float* h      = (const float*)d_in[0];
    const float* coords = (const float*)d_in[1];
    const float* Wq     = (const float*)d_in[2];
    const float* bq     = (const float*)d_in[3];
    const float* Wk     = (const float*)d_in[4];
    const float* bk     = (const float*)d_in[5];
    const float* Wv     = (const float*)d_in[6];
    const float* bv     = (const float*)d_in[7];
    const float* Wd1    = (const float*)d_in[8];
    const float* bd1    = (const float*)d_in[9];
    const float* Wd2    = (const float*)d_in[10];
    const float* bd2    = (const float*)d_in[11];
    const float* Wo     = (const float*)d_in[12];
    const float* bo     = (const float*)d_in[13];
    const float* gamma  = (const float*)d_in[14];
    const float* beta   = (const float*)d_in[15];
    float* out = (float*)d_out;

    const size_t MAT = (size_t)BATCH * NN * HH;    // 524288 floats
    float* ws  = (float*)d_ws;
    float* q    = ws;
    float* kbuf = ws + MAT;
    float* vbuf = ws + 2 * MAT;
    float* ao   = ws + 3 * MAT;
    float* proj = ws + 4 * MAT;

    qkv_gemm_kernel<<<256, 256, 0, stream>>>(h, Wq, bq, Wk, bk, Wv, bv, q, kbuf, vbuf);
    attn_kernel<<<BATCH * (NN / 16), 256, 0, stream>>>(q, kbuf, vbuf, coords,
                                                       Wd1, bd1, Wd2, bd2, ao);
    out_gemm_kernel<<<256, 256, 0, stream>>>(ao, Wo, bo, proj);
    epilogue_kernel<<<512, 256, 0, stream>>>(proj, h, gamma, beta, out);
}